// SlotAttention_LearnableSlots_attnPool_43568148251288
// MI455X (gfx1250) — compile-verified
//
#include <hip/hip_runtime.h>
#include <hip/hip_bf16.h>
#include <math.h>

typedef __attribute__((ext_vector_type(2))) float v2f;
typedef __attribute__((ext_vector_type(8))) float v8f;

#define D_DIM 768
#define NSLOT 8
#define NTOK  1024
#define BATCH 32

#define BM 32
#define BN 64
#define BK 32
#define LDS_STRIDE (BK + 2)

// ---- CDNA5 async global->LDS staging (ASYNCcnt path), with sync fallback ----
#if defined(__has_builtin)
#  if __has_builtin(__builtin_amdgcn_global_load_async_to_lds_b32)
#    define HAVE_ASYNC_LDS 1
#  endif
#endif
#ifndef HAVE_ASYNC_LDS
#  define HAVE_ASYNC_LDS 0
#endif

typedef __attribute__((address_space(1))) int as1_i32;
typedef __attribute__((address_space(3))) int as3_i32;

__device__ __forceinline__ void async_g2l_b32(const float* g, float* l)
{
#if HAVE_ASYNC_LDS
    __builtin_amdgcn_global_load_async_to_lds_b32(
        (as1_i32*)(float*)g,      // global source  (addrspace(1) int*)
        (as3_i32*)l,              // LDS destination (addrspace(3) int*)
        /*imm offset*/ 0, /*cpol*/ 0);
#else
    (void)g; (void)l;
#endif
}

__device__ __forceinline__ void async_wait0()
{
#if HAVE_ASYNC_LDS
#  if defined(__has_builtin) && __has_builtin(__builtin_amdgcn_s_wait_asynccnt)
    __builtin_amdgcn_s_wait_asynccnt(0);
#  else
    asm volatile("s_wait_asynccnt 0x0" ::: "memory");
#  endif
#endif
}

// ---------------------------------------------------------------------------
// Generic fp32 WMMA GEMM:  C[b] = epilogue( alpha * A[b] @ B[b] )
//   TRANSB=0 : B is [K,N] row-major (ldb = row stride)
//   TRANSB=1 : logical B[k][n] = Bsrc[n*ldb + k]  (i.e. A @ Bsrc^T)
//   BIAS / RELU / RESID : epilogue ops
//   GUARD : M is not a multiple of BM -> guard A loads (zero-fill) + C stores
// Block: 256 threads (8 wave32), tile 32x64, K staged in LDS by 32.
// ---------------------------------------------------------------------------
template<bool TRANSB, bool BIAS, bool RELU, bool RESID, bool GUARD>
__global__ __launch_bounds__(256)
void gemm_wmma_f32(const float* __restrict__ A, const float* __restrict__ B,
                   const float* __restrict__ bias, const float* __restrict__ resid,
                   float* __restrict__ C,
                   int M, int N, int K, int lda, int ldb, int ldc,
                   long long sA, long long sB, long long sC, float alpha)
{
    __shared__ __align__(16) float As[BM * LDS_STRIDE];
    __shared__ __align__(16) float Bs[BN * LDS_STRIDE];

    const int t    = threadIdx.x;
    const int wave = t >> 5;
    const int lane = t & 31;
    const int bz   = blockIdx.z;
    const int m0   = blockIdx.y * BM;
    const int n0   = blockIdx.x * BN;

    const float* Ab = A + (long long)bz * sA;
    const float* Bb = B + (long long)bz * sB;
    float*       Cb = C + (long long)bz * sC;
    const float* Rb = RESID ? (resid + (long long)bz * sC) : nullptr;

    const int tr     = wave & 1;   // 16-row slab within the 32-row tile
    const int tc     = wave >> 1;  // 16-col slab within the 64-col tile
    const int laneM  = lane & 15;
    const int laneHi = lane >> 4;  // 0 or 1

    v8f acc = {};

    for (int k0 = 0; k0 < K; k0 += BK) {
        // ---- stage A tile [BM x BK] ----
        if constexpr (!GUARD) {
#if HAVE_ASYNC_LDS
            #pragma unroll
            for (int i = 0; i < 4; ++i) {
                int idx = t + i * 256;
                int mm  = idx >> 5;
                int kk  = idx & 31;
                async_g2l_b32(Ab + (long long)(m0 + mm) * lda + (k0 + kk),
                              &As[mm * LDS_STRIDE + kk]);
            }
#else
            #pragma unroll
            for (int i = 0; i < 4; ++i) {
                int idx = t + i * 256;
                int mm  = idx >> 5;
                int kk  = idx & 31;
                As[mm * LDS_STRIDE + kk] = Ab[(long long)(m0 + mm) * lda + (k0 + kk)];
            }
            if (k0 + BK < K)
                __builtin_prefetch(&Ab[(long long)(m0 + (t >> 3)) * lda + (k0 + BK) + ((t & 7) << 2)], 0, 1);
#endif
        } else {
            // guarded (rows >= M zero-filled) -- tiny batched GEMMs only
            #pragma unroll
            for (int i = 0; i < 4; ++i) {
                int idx = t + i * 256;
                int mm  = idx >> 5;
                int kk  = idx & 31;
                int gm  = m0 + mm;
                float v = 0.0f;
                if (gm < M) v = Ab[(long long)gm * lda + (k0 + kk)];
                As[mm * LDS_STRIDE + kk] = v;
            }
        }
        // ---- stage B tile as Bt[n][k] (never needs guarding: N,K multiples) ----
        #pragma unroll
        for (int i = 0; i < 8; ++i) {
            int idx = t + i * 256;
            int nn, kk;
            const float* src;
            if constexpr (TRANSB) {
                kk  = idx & 31;
                nn  = idx >> 5;
                src = Bb + (long long)(n0 + nn) * ldb + (k0 + kk);
            } else {
                nn  = idx & 63;
                kk  = idx >> 6;
                src = Bb + (long long)(k0 + kk) * ldb + (n0 + nn);
            }
#if HAVE_ASYNC_LDS
            async_g2l_b32(src, &Bs[nn * LDS_STRIDE + kk]);
#else
            Bs[nn * LDS_STRIDE + kk] = *src;
#endif
        }
        async_wait0();
        __syncthreads();

        // fp32 WMMA fragment layout (16x4 A / 4x16 B):
        //   lane's pair for k-step s lives at kk = 4*s + 2*laneHi (contiguous 8B)
        const float* Ar = &As[(tr * 16 + laneM) * LDS_STRIDE];
        const float* Br = &Bs[(tc * 16 + laneM) * LDS_STRIDE];
        #pragma unroll
        for (int s = 0; s < BK / 4; ++s) {
            int kk = (s << 2) + (laneHi << 1);
            v2f av = *(const v2f*)(Ar + kk);
            v2f bv = *(const v2f*)(Br + kk);
            acc = __builtin_amdgcn_wmma_f32_16x16x4_f32(false, av, false, bv,
                                                        (short)0, acc, false, false);
        }
        __syncthreads();
    }

    // ---- epilogue: C/D layout: vgpr r -> row r (lanes 0-15) / r+8 (lanes 16-31)
    #pragma unroll
    for (int r = 0; r < 8; ++r) {
        int row = m0 + tr * 16 + r + (laneHi << 3);
        int col = n0 + tc * 16 + laneM;
        if (!GUARD || row < M) {
            float v = acc[r] * alpha;
            if (BIAS)  v += bias[col];
            if (RESID) v += Rb[(long long)row * ldc + col];
            if (RELU)  v = fmaxf(v, 0.0f);
            Cb[(long long)row * ldc + col] = v;
        }
    }
}

// ---------------------------------------------------------------------------
// LayerNorm over D=768, one block (256 threads) per row, two-pass mean/var.
// ---------------------------------------------------------------------------
__global__ __launch_bounds__(256)
void ln768(const float* __restrict__ in, const float* __restrict__ g,
           const float* __restrict__ be, float* __restrict__ out)
{
    __shared__ float red[256];
    long long row = blockIdx.x;
    const float* x = in + row * D_DIM;
    float*       y = out + row * D_DIM;
    int t = threadIdx.x;
    float v0 = x[t], v1 = x[t + 256], v2 = x[t + 512];
    red[t] = v0 + v1 + v2;
    __syncthreads();
    for (int off = 128; off > 0; off >>= 1) {
        if (t < off) red[t] += red[t + off];
        __syncthreads();
    }
    float mean = red[0] * (1.0f / 768.0f);
    __syncthreads();
    float d0 = v0 - mean, d1 = v1 - mean, d2 = v2 - mean;
    red[t] = d0 * d0 + d1 * d1 + d2 * d2;
    __syncthreads();
    for (int off = 128; off > 0; off >>= 1) {
        if (t < off) red[t] += red[t + off];
        __syncthreads();
    }
    float rs = rsqrtf(red[0] * (1.0f / 768.0f) + 1e-5f);
    y[t]       = d0 * rs * g[t]       + be[t];
    y[t + 256] = d1 * rs * g[t + 256] + be[t + 256];
    y[t + 512] = d2 * rs * g[t + 512] + be[t + 512];
}

// softmax over the 8 slots for each (b, token j); in-place on dots buffer
__global__ void softmax_over_slots(float* __restrict__ attn)
{
    int idx = blockIdx.x * 256 + threadIdx.x;    // 0 .. B*NTOK-1
    int b = idx >> 10;
    int j = idx & 1023;
    float* base = attn + (long long)b * (NSLOT * NTOK) + j;
    float d[NSLOT];
    float mx = -3.0e38f;
    #pragma unroll
    for (int i = 0; i < NSLOT; ++i) { d[i] = base[i * NTOK]; mx = fmaxf(mx, d[i]); }
    float s = 0.0f;
    #pragma unroll
    for (int i = 0; i < NSLOT; ++i) { d[i] = expf(d[i] - mx); s += d[i]; }
    float inv = 1.0f / s;
    #pragma unroll
    for (int i = 0; i < NSLOT; ++i) base[i * NTOK] = d[i] * inv;
}

// attn[b,i,:] /= (sum_j attn[b,i,j] + 1e-8); one block per (b,i) row
__global__ __launch_bounds__(256)
void attn_renorm(float* __restrict__ attn)
{
    __shared__ float red[256];
    float* a = attn + (long long)blockIdx.x * NTOK;
    int t = threadIdx.x;
    float v0 = a[t], v1 = a[t + 256], v2 = a[t + 512], v3 = a[t + 768];
    red[t] = v0 + v1 + v2 + v3;
    __syncthreads();
    for (int off = 128; off > 0; off >>= 1) {
        if (t < off) red[t] += red[t + off];
        __syncthreads();
    }
    float inv = 1.0f / (red[0] + 1e-8f);
    a[t] = v0 * inv; a[t + 256] = v1 * inv; a[t + 512] = v2 * inv; a[t + 768] = v3 * inv;
}

// broadcast learned slots [1,8,768] -> [32,8,768]
__global__ void init_slots(const float* __restrict__ src, float* __restrict__ dst)
{
    int idx = blockIdx.x * 256 + threadIdx.x;    // < 196608
    dst[idx] = src[idx % (NSLOT * D_DIM)];
}

// pack outputs: [slots flat | attn transposed to [b, n, ns]]
__global__ void write_out(const float* __restrict__ slots, const float* __restrict__ attn,
                          float* __restrict__ out)
{
    int idx = blockIdx.x * 256 + threadIdx.x;
    const int n1 = BATCH * NSLOT * D_DIM;   // 196608
    const int n2 = BATCH * NTOK * NSLOT;    // 262144
    if (idx < n1) {
        out[idx] = slots[idx];
    } else if (idx < n1 + n2) {
        int o = idx - n1;
        int b = o >> 13;        // / (NTOK*NSLOT)
        int r = o & 8191;
        int j = r >> 3;
        int i = r & 7;
        out[idx] = attn[(long long)b * (NSLOT * NTOK) + i * NTOK + j];
    }
}

// ---------------------------------------------------------------------------
extern "C" void kernel_launch(void* const* d_in, const int* in_sizes, int n_in,
                              void* d_out, int out_size, void* d_ws, size_t ws_size,
                              hipStream_t stream)
{
    (void)in_sizes; (void)n_in; (void)out_size; (void)ws_size;

    const float* slots0 = (const float*)d_in[0];
    const float* inputs = (const float*)d_in[1];
    const float* Wq = (const float*)d_in[2];
    const float* bq = (const float*)d_in[3];
    const float* Wk = (const float*)d_in[4];
    const float* bk = (const float*)d_in[5];
    const float* Wv = (const float*)d_in[6];
    const float* bv = (const float*)d_in[7];
    const float* g_in = (const float*)d_in[8];
    const float* b_in = (const float*)d_in[9];
    const float* g_s  = (const float*)d_in[10];
    const float* b_s  = (const float*)d_in[11];
    const float* g_ff = (const float*)d_in[12];
    const float* b_ff = (const float*)d_in[13];
    const float* W1 = (const float*)d_in[14];
    const float* b1 = (const float*)d_in[15];
    const float* W2 = (const float*)d_in[16];
    const float* b2 = (const float*)d_in[17];

    const long long MN = (long long)BATCH * NTOK;   // 32768 rows
    float* ws    = (float*)d_ws;
    float* x     = ws;
    float* kbuf  = x    + MN * D_DIM;
    float* vbuf  = kbuf + MN * D_DIM;
    float* slots = vbuf + MN * D_DIM;
    float* sn    = slots + BATCH * NSLOT * D_DIM;
    float* q     = sn    + BATCH * NSLOT * D_DIM;
    float* attn  = q     + BATCH * NSLOT * D_DIM;   // [B, NS, NTOK]
    float* upd   = attn  + BATCH * NSLOT * NTOK;
    float* h     = upd   + BATCH * NSLOT * D_DIM;
    float* ff1   = h     + BATCH * NSLOT * D_DIM;

    const float scale = 1.0f / sqrtf((float)D_DIM);

    // x = LN(inputs)
    ln768<<<dim3((unsigned)MN), 256, 0, stream>>>(inputs, g_in, b_in, x);

    // k = x@Wk + bk ; v = x@Wv + bv   (the 77 GFLOP bulk; unguarded fast path)
    gemm_wmma_f32<false, true, false, false, false><<<dim3(D_DIM / BN, MN / BM, 1), 256, 0, stream>>>(
        x, Wk, bk, nullptr, kbuf, (int)MN, D_DIM, D_DIM, D_DIM, D_DIM, D_DIM, 0, 0, 0, 1.0f);
    gemm_wmma_f32<false, true, false, false, false><<<dim3(D_DIM / BN, MN / BM, 1), 256, 0, stream>>>(
        x, Wv, bv, nullptr, vbuf, (int)MN, D_DIM, D_DIM, D_DIM, D_DIM, D_DIM, 0, 0, 0, 1.0f);

    init_slots<<<(BATCH * NSLOT * D_DIM) / 256, 256, 0, stream>>>(slots0, slots);

    for (int it = 0; it < 3; ++it) {
        // sn = LN(slots); q = sn@Wq + bq
        ln768<<<BATCH * NSLOT, 256, 0, stream>>>(slots, g_s, b_s, sn);
        gemm_wmma_f32<false, true, false, false, false><<<dim3(D_DIM / BN, (BATCH * NSLOT) / BM, 1), 256, 0, stream>>>(
            sn, Wq, bq, nullptr, q, BATCH * NSLOT, D_DIM, D_DIM, D_DIM, D_DIM, D_DIM, 0, 0, 0, 1.0f);

        // dots[b] = scale * q[b] @ k[b]^T   -> attn buffer [8,1024] per batch (M=8: guarded)
        gemm_wmma_f32<true, false, false, false, true><<<dim3(NTOK / BN, 1, BATCH), 256, 0, stream>>>(
            q, kbuf, nullptr, nullptr, attn, NSLOT, NTOK, D_DIM, D_DIM, D_DIM, NTOK,
            (long long)NSLOT * D_DIM, (long long)NTOK * D_DIM, (long long)NSLOT * NTOK, scale);

        softmax_over_slots<<<(BATCH * NTOK) / 256, 256, 0, stream>>>(attn);
        attn_renorm<<<BATCH * NSLOT, 256, 0, stream>>>(attn);

        // updates[b] = attn[b] @ v[b]   (M=8: guarded)
        gemm_wmma_f32<false, false, false, false, true><<<dim3(D_DIM / BN, 1, BATCH), 256, 0, stream>>>(
            attn, vbuf, nullptr, nullptr, upd, NSLOT, D_DIM, NTOK, NTOK, D_DIM, D_DIM,
            (long long)NSLOT * NTOK, (long long)NTOK * D_DIM, (long long)NSLOT * D_DIM, 1.0f);

        // h = LN(updates); slots += relu(h@W1+b1)@W2 + b2
        ln768<<<BATCH * NSLOT, 256, 0, stream>>>(upd, g_ff, b_ff, h);
        gemm_wmma_f32<false, true, true, false, false><<<dim3(D_DIM / BN, (BATCH * NSLOT) / BM, 1), 256, 0, stream>>>(
            h, W1, b1, nullptr, ff1, BATCH * NSLOT, D_DIM, D_DIM, D_DIM, D_DIM, D_DIM, 0, 0, 0, 1.0f);
        gemm_wmma_f32<false, true, false, true, false><<<dim3(D_DIM / BN, (BATCH * NSLOT) / BM, 1), 256, 0, stream>>>(
            ff1, W2, b2, slots, slots, BATCH * NSLOT, D_DIM, D_DIM, D_DIM, D_DIM, D_DIM, 0, 0, 0, 1.0f);
    }

    int tot = BATCH * NSLOT * D_DIM + BATCH * NTOK * NSLOT;
    write_out<<<(tot + 255) / 256, 256, 0, stream>>>(slots, attn, (float*)d_out);
}